// Recommender_35837207118176
// MI455X (gfx1250) — compile-verified
//
#include <hip/hip_runtime.h>

typedef float v2f __attribute__((ext_vector_type(2)));
typedef float v8f __attribute__((ext_vector_type(8)));

namespace {
constexpr int       kUsers = 100000;
constexpr int       kItems = 50000;
constexpr int       kEnt   = 120000;
constexpr int       kDim   = 64;
constexpr int       kEdges = 1000000;
constexpr int       kCF    = 1000000;
constexpr float     kGamma = 0.6f;
constexpr long long kOutEnt  = 0;
constexpr long long kOutUser = (long long)kEnt * kDim;                      // 7,680,000
constexpr long long kOutUcf  = kOutUser + (long long)kUsers * kDim;         // 14,080,000
constexpr long long kOutIcf  = kOutUcf + (long long)kUsers * kDim;          // 20,480,000
}

__device__ __forceinline__ float sigmoidf(float x) { return 1.0f / (1.0f + expf(-x)); }
__device__ __forceinline__ float lrelu(float x) { return x >= 0.0f ? x : 0.01f * x; }
__device__ __forceinline__ float4 ld4(const float* p) { return *reinterpret_cast<const float4*>(p); }

// ---------------- utility kernels ----------------
__global__ void k_zero(float* __restrict__ p, long long n) {
  long long i = (long long)blockIdx.x * blockDim.x + threadIdx.x;
  long long st = (long long)gridDim.x * blockDim.x;
  for (; i < n; i += st) p[i] = 0.0f;
}

__global__ void k_copy2(const float* __restrict__ src, float* __restrict__ d1,
                        float* __restrict__ d2, long long n) {
  long long i = (long long)blockIdx.x * blockDim.x + threadIdx.x;
  long long st = (long long)gridDim.x * blockDim.x;
  for (; i < n; i += st) { float v = src[i]; d1[i] = v; d2[i] = v; }
}

// ---------------- KG edge scatter (seg_mean_w numerators + counts) ----------------
__global__ void k_edge_scatter(const int* __restrict__ ei, const int* __restrict__ et,
                               const float* __restrict__ relw, const float* __restrict__ ent,
                               float* __restrict__ agg1, float* __restrict__ agg2,
                               float* __restrict__ relsum, float* __restrict__ c1,
                               float* __restrict__ c2, float* __restrict__ crel) {
  long long total = (long long)kEdges * 16;
  long long i = (long long)blockIdx.x * blockDim.x + threadIdx.x;
  long long st = (long long)gridDim.x * blockDim.x;
  for (; i < total; i += st) {
    int e = (int)(i >> 4), l = (int)(i & 15), d = l * 4;
    int head = ei[e], tail = ei[kEdges + e], t = et[e];
    float4 te = ld4(ent + (long long)tail * kDim + d);
    float4 re = ld4(relw + (long long)t * kDim + d);
    bool cross = (head < kItems) != (tail < kItems);
    long long ho = (long long)head * kDim + d;
    if (cross) {
      atomicAdd(agg1 + ho + 0, te.x * re.x); atomicAdd(agg1 + ho + 1, te.y * re.y);
      atomicAdd(agg1 + ho + 2, te.z * re.z); atomicAdd(agg1 + ho + 3, te.w * re.w);
    } else {
      atomicAdd(agg2 + ho + 0, te.x + re.x); atomicAdd(agg2 + ho + 1, te.y + re.y);
      atomicAdd(agg2 + ho + 2, te.z + re.z); atomicAdd(agg2 + ho + 3, te.w + re.w);
    }
    if (head < kItems) {
      atomicAdd(relsum + ho + 0, re.x); atomicAdd(relsum + ho + 1, re.y);
      atomicAdd(relsum + ho + 2, re.z); atomicAdd(relsum + ho + 3, re.w);
    }
    if (l == 0) {
      if (cross) atomicAdd(c1 + head, 1.0f); else atomicAdd(c2 + head, 1.0f);
      if (head < kItems) atomicAdd(crel + head, 1.0f);
    }
  }
}

// ---------------- fused dual GEMM via fp32 WMMA ----------------
// ent_next = lrelu((agg1/max(c1,1)) @ W1.T + b1)*0.5 + lrelu((agg2/max(c2,1)) @ W2.T + b2)*0.5
__global__ void __launch_bounds__(256)
k_gemm_fused(const float* __restrict__ agg1, const float* __restrict__ agg2,
             const float* __restrict__ c1, const float* __restrict__ c2,
             const float* __restrict__ w1, const float* __restrict__ b1,
             const float* __restrict__ w2, const float* __restrict__ b2,
             float* __restrict__ outp) {
  int wave = threadIdx.x >> 5;
  int lane = threadIdx.x & 31;
  int tile = blockIdx.x * (blockDim.x >> 5) + wave;
  if (tile >= kEnt / 16) return;
  int m = lane & 15, hi = lane >> 4;
  int row = tile * 16 + m;
  float inv1 = 1.0f / fmaxf(c1[row], 1.0f);
  float inv2 = 1.0f / fmaxf(c2[row], 1.0f);
  const float* a1p = agg1 + (long long)row * kDim + hi * 2;
  const float* a2p = agg2 + (long long)row * kDim + hi * 2;
  v8f z = {0.f, 0.f, 0.f, 0.f, 0.f, 0.f, 0.f, 0.f};
  v8f acc1[4], acc2[4];
#pragma unroll
  for (int nt = 0; nt < 4; ++nt) { acc1[nt] = z; acc2[nt] = z; }
  for (int kk = 0; kk < 16; ++kk) {
    int k0 = kk * 4;
    v2f a1 = *reinterpret_cast<const v2f*>(a1p + k0);
    v2f a2 = *reinterpret_cast<const v2f*>(a2p + k0);
    a1[0] *= inv1; a1[1] *= inv1;
    a2[0] *= inv2; a2[1] *= inv2;
#pragma unroll
    for (int nt = 0; nt < 4; ++nt) {
      int n = nt * 16 + m;
      v2f bb1 = *reinterpret_cast<const v2f*>(w1 + (long long)n * kDim + k0 + hi * 2);
      v2f bb2 = *reinterpret_cast<const v2f*>(w2 + (long long)n * kDim + k0 + hi * 2);
      acc1[nt] = __builtin_amdgcn_wmma_f32_16x16x4_f32(false, a1, false, bb1, (short)0,
                                                       acc1[nt], false, false);
      acc2[nt] = __builtin_amdgcn_wmma_f32_16x16x4_f32(false, a2, false, bb2, (short)0,
                                                       acc2[nt], false, false);
    }
  }
#pragma unroll
  for (int nt = 0; nt < 4; ++nt) {
    int n = nt * 16 + m;
    float bias1 = b1[n], bias2 = b2[n];
#pragma unroll
    for (int v = 0; v < 8; ++v) {
      int r = tile * 16 + v + 8 * hi;
      float x1 = lrelu(acc1[nt][v] + bias1) * 0.5f;
      float x2 = lrelu(acc2[nt][v] + bias2) * 0.5f;
      outp[(long long)r * kDim + n] = x1 + x2;
    }
  }
}

// rel_i = relsum / max(crel,1)
__global__ void k_div_rows(float* __restrict__ x, const float* __restrict__ cnt, long long n) {
  long long i = (long long)blockIdx.x * blockDim.x + threadIdx.x;
  long long st = (long long)gridDim.x * blockDim.x;
  for (; i < n; i += st) x[i] /= fmaxf(cnt[i >> 6], 1.0f);
}

// ---------------- CF loop kernels ----------------
__global__ void k_cf_dot(const int* __restrict__ imat, const float* __restrict__ u,
                         const float* __restrict__ ucf, const float* __restrict__ rel,
                         const float* __restrict__ kg, const float* __restrict__ icf,
                         float* __restrict__ p_raw, float* __restrict__ pcf_raw) {
  long long total = (long long)kCF * 16;
  long long i = (long long)blockIdx.x * blockDim.x + threadIdx.x;
  long long st = (long long)gridDim.x * blockDim.x;
  for (; i < total; i += st) {
    int e = (int)(i >> 4), l = (int)(i & 15), d = l * 4;
    int row = imat[2 * e], col = imat[2 * e + 1];
    float4 uu = ld4(u + (long long)row * kDim + d);
    float4 ri = ld4(rel + (long long)col * kDim + d);
    float4 kk = ld4(kg + (long long)col * kDim + d);
    float s = uu.x * ri.x * kk.x + uu.y * ri.y * kk.y + uu.z * ri.z * kk.z + uu.w * ri.w * kk.w;
    float4 uc = ld4(ucf + (long long)row * kDim + d);
    float4 ic = ld4(icf + (long long)col * kDim + d);
    float sc = uc.x * ic.x + uc.y * ic.y + uc.z * ic.z + uc.w * ic.w;
    for (int off = 8; off > 0; off >>= 1) {
      s += __shfl_xor(s, off, 32);
      sc += __shfl_xor(sc, off, 32);
    }
    if (l == 0) { p_raw[e] = sigmoidf(s); pcf_raw[e] = sigmoidf(sc); }
  }
}

__global__ void k_seg_max(const int* __restrict__ imat, const float* __restrict__ p,
                          const float* __restrict__ pcf, float* __restrict__ mxp,
                          float* __restrict__ mxc) {
  long long i = (long long)blockIdx.x * blockDim.x + threadIdx.x;
  long long st = (long long)gridDim.x * blockDim.x;
  for (; i < kCF; i += st) {
    int row = imat[2 * i];
    // values are sigmoid outputs (>= 0): uint bit ordering == float ordering
    atomicMax(reinterpret_cast<unsigned int*>(mxp + row), __float_as_uint(p[i]));
    atomicMax(reinterpret_cast<unsigned int*>(mxc + row), __float_as_uint(pcf[i]));
  }
}

__global__ void k_seg_expsum(const int* __restrict__ imat, const float* __restrict__ p,
                             const float* __restrict__ pcf, const float* __restrict__ mxp,
                             const float* __restrict__ mxc, float* __restrict__ smp,
                             float* __restrict__ smc) {
  long long i = (long long)blockIdx.x * blockDim.x + threadIdx.x;
  long long st = (long long)gridDim.x * blockDim.x;
  for (; i < kCF; i += st) {
    int row = imat[2 * i];
    atomicAdd(smp + row, expf(p[i] - mxp[row]));
    atomicAdd(smc + row, expf(pcf[i] - mxc[row]));
  }
}

__global__ void k_cf_accum(const int* __restrict__ imat, const float* __restrict__ p_raw,
                           const float* __restrict__ pcf_raw, const float* __restrict__ mxp,
                           const float* __restrict__ smp, const float* __restrict__ mxc,
                           const float* __restrict__ smc, const float* __restrict__ kg,
                           const float* __restrict__ icf, float* __restrict__ u_next,
                           float* __restrict__ c_next) {
  long long total = (long long)kCF * 16;
  long long i = (long long)blockIdx.x * blockDim.x + threadIdx.x;
  long long st = (long long)gridDim.x * blockDim.x;
  for (; i < total; i += st) {
    int e = (int)(i >> 4), l = (int)(i & 15), d = l * 4;
    int row = imat[2 * e], col = imat[2 * e + 1];
    float pe = expf(p_raw[e] - mxp[row]) / smp[row];
    float pc = expf(pcf_raw[e] - mxc[row]) / smc[row];
    if (fabsf(sigmoidf(pe) - sigmoidf(pc)) < kGamma) {
      float4 kk = ld4(kg + (long long)col * kDim + d);
      float4 ic = ld4(icf + (long long)col * kDim + d);
      long long uo = (long long)row * kDim + d;
      atomicAdd(u_next + uo + 0, kk.x * pe); atomicAdd(u_next + uo + 1, kk.y * pe);
      atomicAdd(u_next + uo + 2, kk.z * pe); atomicAdd(u_next + uo + 3, kk.w * pe);
      atomicAdd(c_next + uo + 0, ic.x * pc); atomicAdd(c_next + uo + 1, ic.y * pc);
      atomicAdd(c_next + uo + 2, ic.z * pc); atomicAdd(c_next + uo + 3, ic.w * pc);
    }
  }
}

// item_agg numerator (count divide is absorbed by the subsequent row-norm)
__global__ void k_item_agg(const int* __restrict__ imat, const float* __restrict__ ucf0,
                           float* __restrict__ icf_next) {
  long long total = (long long)kCF * 16;
  long long i = (long long)blockIdx.x * blockDim.x + threadIdx.x;
  long long st = (long long)gridDim.x * blockDim.x;
  for (; i < total; i += st) {
    int e = (int)(i >> 4), l = (int)(i & 15), d = l * 4;
    int row = imat[2 * e], col = imat[2 * e + 1];
    float4 v = ld4(ucf0 + (long long)row * kDim + d);
    long long io = (long long)col * kDim + d;
    atomicAdd(icf_next + io + 0, v.x); atomicAdd(icf_next + io + 1, v.y);
    atomicAdd(icf_next + io + 2, v.z); atomicAdd(icf_next + io + 3, v.w);
  }
}

// ---------------- row normalization ----------------
__global__ void k_norm_rows(const float* __restrict__ src, float* __restrict__ dst,
                            long long nrows) {
  long long total = nrows * 16;
  long long i = (long long)blockIdx.x * blockDim.x + threadIdx.x;
  long long st = (long long)gridDim.x * blockDim.x;
  for (; i < total; i += st) {
    long long r = i >> 4;
    int l = (int)(i & 15), d = l * 4;
    float4 v = ld4(src + r * kDim + d);
    float s = v.x * v.x + v.y * v.y + v.z * v.z + v.w * v.w;
    for (int off = 8; off > 0; off >>= 1) s += __shfl_xor(s, off, 32);
    float inv = 1.0f / fmaxf(sqrtf(s), 1e-12f);
    float* dp = dst + r * kDim + d;
    dp[0] = v.x * inv; dp[1] = v.y * inv; dp[2] = v.z * inv; dp[3] = v.w * inv;
  }
}

// normalize + write into cur buffer + accumulate into output residual
__global__ void k_norm_rows_acc(const float* __restrict__ src, float* __restrict__ cur,
                                float* __restrict__ outacc, long long nrows) {
  long long total = nrows * 16;
  long long i = (long long)blockIdx.x * blockDim.x + threadIdx.x;
  long long st = (long long)gridDim.x * blockDim.x;
  for (; i < total; i += st) {
    long long r = i >> 4;
    int l = (int)(i & 15), d = l * 4;
    float4 v = ld4(src + r * kDim + d);
    float s = v.x * v.x + v.y * v.y + v.z * v.z + v.w * v.w;
    for (int off = 8; off > 0; off >>= 1) s += __shfl_xor(s, off, 32);
    float inv = 1.0f / fmaxf(sqrtf(s), 1e-12f);
    long long o = r * kDim + d;
    float n0 = v.x * inv, n1 = v.y * inv, n2 = v.z * inv, n3 = v.w * inv;
    cur[o + 0] = n0; cur[o + 1] = n1; cur[o + 2] = n2; cur[o + 3] = n3;
    outacc[o + 0] += n0; outacc[o + 1] += n1; outacc[o + 2] += n2; outacc[o + 3] += n3;
  }
}

// ---------------- host driver ----------------
static inline dim3 NB(long long n) { return dim3((unsigned)((n + 255) / 256)); }

extern "C" void kernel_launch(void* const* d_in, const int* in_sizes, int n_in,
                              void* d_out, int out_size, void* d_ws, size_t ws_size,
                              hipStream_t stream) {
  (void)in_sizes; (void)n_in; (void)out_size;
  const float* user_emb   = (const float*)d_in[0];
  const float* entity_emb = (const float*)d_in[1];
  const float* emb_cf     = (const float*)d_in[2];
  const float* relw       = (const float*)d_in[3];
  const float* W1w        = (const float*)d_in[4];
  const float* W1b        = (const float*)d_in[5];
  const float* W2w        = (const float*)d_in[6];
  const float* W2b        = (const float*)d_in[7];
  const int*   ei         = (const int*)d_in[8];
  const int*   et         = (const int*)d_in[9];
  const int*   imat       = (const int*)d_in[10];
  float* out = (float*)d_out;

  float* ws = (float*)d_ws;
  size_t o = 0;
  auto take = [&](size_t n) { float* p = ws + o; o += n; return p; };
  float* ent_cur  = take((size_t)kEnt * kDim);
  float* ent_next = take((size_t)kEnt * kDim);
  float* user_cur = take((size_t)kUsers * kDim);
  float* ucf_cur  = take((size_t)kUsers * kDim);
  float* icf_cur  = take((size_t)kItems * kDim);
  float* icf_next = take((size_t)kItems * kDim);
  float* agg1     = take((size_t)kEnt * kDim);   // reused as uA after GEMM
  float* agg2     = take((size_t)kEnt * kDim);   // reused as uB after GEMM
  float* cA       = take((size_t)kUsers * kDim);
  float* cB       = take((size_t)kUsers * kDim);
  float* rel      = take((size_t)kItems * kDim);
  float* c1       = take(kEnt);
  float* c2       = take(kEnt);     // c1,c2,crel contiguous
  float* crel     = take(kItems);
  float* p_raw    = take(kCF);
  float* pcf_raw  = take(kCF);
  float* mxp      = take(kUsers);   // mxp,smp,mxc,smc contiguous
  float* smp      = take(kUsers);
  float* mxc      = take(kUsers);
  float* smc      = take(kUsers);
  if (o * sizeof(float) > ws_size) return;  // insufficient scratch
  float* uA = agg1;
  float* uB = agg2;

  // init output residual accumulators and current-state buffers
  k_copy2<<<NB((long long)kEnt * kDim), 256, 0, stream>>>(entity_emb, ent_cur, out + kOutEnt,
                                                          (long long)kEnt * kDim);
  k_copy2<<<NB((long long)kUsers * kDim), 256, 0, stream>>>(user_emb, user_cur, out + kOutUser,
                                                            (long long)kUsers * kDim);
  k_copy2<<<NB((long long)kUsers * kDim), 256, 0, stream>>>(emb_cf, ucf_cur, out + kOutUcf,
                                                            (long long)kUsers * kDim);
  k_copy2<<<NB((long long)kItems * kDim), 256, 0, stream>>>(emb_cf + (size_t)kUsers * kDim,
                                                            icf_cur, out + kOutIcf,
                                                            (long long)kItems * kDim);

  for (int h = 0; h < 2; ++h) {
    // --- KG aggregation ---
    k_zero<<<NB(2LL * kEnt * kDim), 256, 0, stream>>>(agg1, 2LL * kEnt * kDim);
    k_zero<<<NB((long long)kItems * kDim), 256, 0, stream>>>(rel, (long long)kItems * kDim);
    k_zero<<<NB(2LL * kEnt + kItems), 256, 0, stream>>>(c1, 2LL * kEnt + kItems);
    k_edge_scatter<<<NB(16LL * kEdges), 256, 0, stream>>>(ei, et, relw, ent_cur, agg1, agg2,
                                                          rel, c1, c2, crel);
    k_gemm_fused<<<dim3((kEnt / 16 + 7) / 8), 256, 0, stream>>>(
        agg1, agg2, c1, c2, W1w + (size_t)h * kDim * kDim, W1b + (size_t)h * kDim,
        W2w + (size_t)h * kDim * kDim, W2b + (size_t)h * kDim, ent_next);
    k_div_rows<<<NB((long long)kItems * kDim), 256, 0, stream>>>(rel, crel,
                                                                 (long long)kItems * kDim);

    // --- CF denoising loop (3 iterations, ping-pong) ---
    const float* u_in = user_cur;
    const float* c_in = ucf_cur;
    float* u_out = uA;
    float* c_out = cA;
    for (int it = 0; it < 3; ++it) {
      k_zero<<<NB(4LL * kUsers), 256, 0, stream>>>(mxp, 4LL * kUsers);
      k_cf_dot<<<NB(16LL * kCF), 256, 0, stream>>>(imat, u_in, c_in, rel, ent_cur, icf_cur,
                                                   p_raw, pcf_raw);
      k_seg_max<<<NB(kCF), 256, 0, stream>>>(imat, p_raw, pcf_raw, mxp, mxc);
      k_seg_expsum<<<NB(kCF), 256, 0, stream>>>(imat, p_raw, pcf_raw, mxp, mxc, smp, smc);
      k_zero<<<NB((long long)kUsers * kDim), 256, 0, stream>>>(u_out, (long long)kUsers * kDim);
      k_zero<<<NB((long long)kUsers * kDim), 256, 0, stream>>>(c_out, (long long)kUsers * kDim);
      k_cf_accum<<<NB(16LL * kCF), 256, 0, stream>>>(imat, p_raw, pcf_raw, mxp, smp, mxc, smc,
                                                     ent_cur, icf_cur, u_out, c_out);
      if (it < 2) {
        k_norm_rows<<<NB(16LL * kUsers), 256, 0, stream>>>(u_out, u_out, kUsers);
        k_norm_rows<<<NB(16LL * kUsers), 256, 0, stream>>>(c_out, c_out, kUsers);
      }
      u_in = u_out; c_in = c_out;
      u_out = (u_out == uA) ? uB : uA;
      c_out = (c_out == cA) ? cB : cA;
    }

    // --- item aggregation from hop-input user_cf ---
    k_zero<<<NB((long long)kItems * kDim), 256, 0, stream>>>(icf_next,
                                                             (long long)kItems * kDim);
    k_item_agg<<<NB(16LL * kCF), 256, 0, stream>>>(imat, ucf_cur, icf_next);

    // --- normalize, feed next hop, accumulate residual output ---
    k_norm_rows_acc<<<NB(16LL * kEnt), 256, 0, stream>>>(ent_next, ent_cur, out + kOutEnt, kEnt);
    k_norm_rows_acc<<<NB(16LL * kUsers), 256, 0, stream>>>(u_in, user_cur, out + kOutUser,
                                                           kUsers);
    k_norm_rows_acc<<<NB(16LL * kUsers), 256, 0, stream>>>(c_in, ucf_cur, out + kOutUcf, kUsers);
    k_norm_rows_acc<<<NB(16LL * kItems), 256, 0, stream>>>(icf_next, icf_cur, out + kOutIcf,
                                                           kItems);
  }
}